// AFNO1D_Mixing_7086696038657
// MI455X (gfx1250) — compile-verified
//
#include <hip/hip_runtime.h>
#include <math.h>

#define HIDDEN   768
#define NB       8
#define BLK      96
#define SEQ      8192
#define N2       4096          // half-size complex FFT length
#define MPAD     4112          // 257*16, covers 4097 modes + zero pad
#define BATCH    4
#define LAMBDA_  0.01f
#define NKC      24            // 96 / 4 K-chunks for 16x16x4 WMMA

typedef __attribute__((ext_vector_type(2))) float v2f;
typedef __attribute__((ext_vector_type(8))) float v8f;

// ---------------------------------------------------------------------------
// fp32 WMMA: D = A(16x4) * B(4x16) + C(16x16), wave32.
// A layout (ISA 7.12.2): lane = 16*khalf + row holds A[row][4kc+2*khalf + {0,1}]
// B layout (symmetric):  lane = 16*khalf + col holds B[4kc+2*khalf + {0,1}][col]
// C/D layout: lanes 0-15 col N=lane rows M=v; lanes 16-31 rows M=8+v.
// fp32 WMMA has no A/B NEG (CNeg only) -> negate fragments in VALU instead.
// ---------------------------------------------------------------------------
__device__ __forceinline__ v8f wmma4(v2f a, v2f b, v8f c) {
  return __builtin_amdgcn_wmma_f32_16x16x4_f32(false, a, false, b, (short)0, c,
                                               false, false);
}

__device__ __forceinline__ float sshrink(float v) {
  float t = fabsf(v) - LAMBDA_;
  return t > 0.0f ? copysignf(t, v) : 0.0f;
}

// ---------------------------------------------------------------------------
// In-LDS 4096-point complex Stockham radix-2 FFT, 12 stages, 256 threads.
// sign = -1 forward DFT, +1 unnormalized inverse DFT. Result ends in bufA.
// ---------------------------------------------------------------------------
__device__ void fft4096(float2* bufA, float2* bufB, int tid, float sign) {
  float2* src = bufA;
  float2* dst = bufB;
  int mh = N2 >> 1;  // 2048
#pragma unroll 1
  for (int stage = 0; stage < 12; ++stage) {
    __syncthreads();
    const int s = 1 << stage;
    for (int i = tid; i < N2 / 2; i += 256) {
      const int p = i >> stage;
      const int q = i & (s - 1);
      float2 a = src[q + s * p];
      float2 b = src[q + s * (p + mh)];
      float th = sign * 3.14159265358979323846f * (float)p / (float)mh;
      float sw, cw;
      __sincosf(th, &sw, &cw);
      float2 sum;
      sum.x = a.x + b.x;
      sum.y = a.y + b.y;
      const float dx = a.x - b.x, dy = a.y - b.y;
      float2 tw;
      tw.x = dx * cw - dy * sw;
      tw.y = dx * sw + dy * cw;
      dst[q + s * (2 * p)]     = sum;
      dst[q + s * (2 * p + 1)] = tw;
    }
    float2* t = src; src = dst; dst = t;
    mh >>= 1;
  }
  __syncthreads();  // 12 swaps (even) -> result is in bufA
}

// ---------------------------------------------------------------------------
// Kernel 1: x[B,S,C] -> xT[B,C,S]   (coalesced tiled transpose)
// ---------------------------------------------------------------------------
__global__ __launch_bounds__(256) void k_transpose_in(
    const float* __restrict__ x, float* __restrict__ xT) {
  __shared__ float tile[32][33];
  const int b  = blockIdx.z;
  const int s0 = blockIdx.x * 32;
  const int c0 = blockIdx.y * 32;
  const int tx = threadIdx.x & 31;
  const int ty = threadIdx.x >> 5;  // 0..7
#pragma unroll
  for (int j = 0; j < 4; ++j) {
    const int s = s0 + ty + j * 8;
    tile[ty + j * 8][tx] = x[((size_t)b * SEQ + s) * HIDDEN + (c0 + tx)];
  }
  __syncthreads();
#pragma unroll
  for (int j = 0; j < 4; ++j) {
    const int c = c0 + ty + j * 8;
    xT[((size_t)b * HIDDEN + c) * SEQ + (s0 + tx)] = tile[tx][ty + j * 8];
  }
}

// ---------------------------------------------------------------------------
// Kernel 2: forward rfft (ortho) per (b,c) row. 8192-pt real FFT via 4096-pt
// complex FFT on z[t] = x[2t] + i x[2t+1], then even/odd recombination.
// Writes Xr/Xi[b*C+c][0..4096], zero pad to MPAD.
// ---------------------------------------------------------------------------
__global__ __launch_bounds__(256) void k_fft_fwd(
    const float* __restrict__ xT, float* __restrict__ Xr,
    float* __restrict__ Xi) {
  __shared__ float2 bufA[N2];
  __shared__ float2 bufB[N2];
  const int bc  = blockIdx.x;  // b*HIDDEN + c
  const int tid = threadIdx.x;
  const float2* rp2 = (const float2*)(xT + (size_t)bc * SEQ);
  for (int i = tid; i < N2; i += 256) bufA[i] = rp2[i];

  fft4096(bufA, bufB, tid, -1.0f);

  const float sn = 0.011048543456039806f;  // 1/sqrt(8192)
  const size_t ob = (size_t)bc * MPAD;
  for (int k = tid; k <= N2 / 2; k += 256) {
    float2 Zk = bufA[k];
    float2 Zm = bufA[(N2 - k) & (N2 - 1)];
    // Xe = (Zk + conj(Zm))/2 ; Xo = (Zk - conj(Zm))/(2i)
    const float xer = 0.5f * (Zk.x + Zm.x);
    const float xei = 0.5f * (Zk.y - Zm.y);
    const float dr  = 0.5f * (Zk.x - Zm.x);
    const float di  = 0.5f * (Zk.y + Zm.y);
    const float xor_ = di, xoi = -dr;
    float th = -6.283185307179586f * (float)k / (float)SEQ;
    float sw, cw;
    __sincosf(th, &sw, &cw);
    const float wxr = cw * xor_ - sw * xoi;
    const float wxi = cw * xoi + sw * xor_;
    Xr[ob + k] = sn * (xer + wxr);
    Xi[ob + k] = sn * (xei + wxi);
    Xr[ob + N2 - k] = sn * (xer - wxr);   // X[4096-k] = sn*conj(Xe - w*Xo)
    Xi[ob + N2 - k] = -sn * (xei - wxi);
  }
  for (int k = N2 + 1 + tid; k < MPAD; k += 256) {
    Xr[ob + k] = 0.0f;
    Xi[ob + k] = 0.0f;
  }
}

// ---------------------------------------------------------------------------
// Kernel 3: block-diagonal complex 2-layer MLP in frequency domain (WMMA).
// grid = (65 mtile-groups, NB, BATCH); block = 128 (4 waves, 1 tile each).
// Each wave: 16 modes x 96 channels of block n, both layers fused; writes
// result in-place over Xr/Xi.
// ---------------------------------------------------------------------------
__global__ __launch_bounds__(128) void k_afno_mlp(
    float* __restrict__ Xr, float* __restrict__ Xi,
    const float* __restrict__ w1, const float* __restrict__ b1,
    const float* __restrict__ w2, const float* __restrict__ b2) {
  __shared__ float sRbuf[4][16 * 97];
  __shared__ float sIbuf[4][16 * 97];
  const int lane  = threadIdx.x & 31;
  const int wave  = threadIdx.x >> 5;
  const int n     = blockIdx.y;
  const int b     = blockIdx.z;
  const int mtile = blockIdx.x * 4 + wave;
  if (mtile >= MPAD / 16) return;  // wave-uniform; EXEC stays full for WMMA
  const int m0    = mtile * 16;
  const int row   = lane & 15;   // A-row (mode) / B-col (out chan) / D-col
  const int khalf = lane >> 4;   // selects K pair {0,1} or {2,3}

  const size_t xbase = ((size_t)b * HIDDEN + n * BLK) * MPAD;
  const float* w1r = w1 + (size_t)(0 * NB + n) * BLK * BLK;
  const float* w1i = w1 + (size_t)(1 * NB + n) * BLK * BLK;
  const float* w2r = w2 + (size_t)(0 * NB + n) * BLK * BLK;
  const float* w2i = w2 + (size_t)(1 * NB + n) * BLK * BLK;
  const float* b1r = b1 + n * BLK;
  const float* b1i = b1 + NB * BLK + n * BLK;
  const float* b2r = b2 + n * BLK;
  const float* b2i = b2 + NB * BLK + n * BLK;

  // ---- load A fragments of the input spectrum tile (16 modes x 96) ----
  v2f ar[NKC], ai[NKC];
  for (int kc = 0; kc < NKC; ++kc) {
    const int d0 = 4 * kc + 2 * khalf;
    const float* pr = Xr + xbase + (size_t)d0 * MPAD + m0 + row;
    const float* pi = Xi + xbase + (size_t)d0 * MPAD + m0 + row;
    ar[kc][0] = pr[0];    ar[kc][1] = pr[MPAD];
    ai[kc][0] = pi[0];    ai[kc][1] = pi[MPAD];
  }

  float* sR = sRbuf[wave];
  float* sI = sIbuf[wave];

  // ---- layer 1: o1r = relu(xr*W1r - xi*W1i + b1r); o1i = relu(xi*W1r + xr*W1i + b1i)
  for (int j = 0; j < 6; ++j) {
    const float br = b1r[16 * j + row];
    const float bi = b1i[16 * j + row];
    v8f cr, ci;
#pragma unroll
    for (int v = 0; v < 8; ++v) { cr[v] = br; ci[v] = bi; }
    for (int kc = 0; kc < NKC; ++kc) {
      const int d0 = 4 * kc + 2 * khalf;
      v2f wr, wi;
      wr[0] = w1r[d0 * BLK + 16 * j + row];
      wr[1] = w1r[(d0 + 1) * BLK + 16 * j + row];
      wi[0] = w1i[d0 * BLK + 16 * j + row];
      wi[1] = w1i[(d0 + 1) * BLK + 16 * j + row];
      cr = wmma4(ar[kc], wr, cr);
      cr = wmma4(-ai[kc], wi, cr);   // subtraction via negated A fragment
      ci = wmma4(ai[kc], wr, ci);
      ci = wmma4(ar[kc], wi, ci);
    }
#pragma unroll
    for (int v = 0; v < 8; ++v) {
      const int M = khalf * 8 + v;
      sR[M * 97 + 16 * j + row] = fmaxf(cr[v], 0.0f);
      sI[M * 97 + 16 * j + row] = fmaxf(ci[v], 0.0f);
    }
  }

  // ---- reload o1 tile as A fragments (wave-private LDS, in-order DS pipe) ----
  v2f a2r[NKC], a2i[NKC];
  for (int kc = 0; kc < NKC; ++kc) {
    const int d0 = 4 * kc + 2 * khalf;
    a2r[kc][0] = sR[row * 97 + d0];
    a2r[kc][1] = sR[row * 97 + d0 + 1];
    a2i[kc][0] = sI[row * 97 + d0];
    a2i[kc][1] = sI[row * 97 + d0 + 1];
  }

  // ---- layer 2 + softshrink; stage through LDS, then coalesced store ----
  for (int j = 0; j < 6; ++j) {
    const float br = b2r[16 * j + row];
    const float bi = b2i[16 * j + row];
    v8f cr, ci;
#pragma unroll
    for (int v = 0; v < 8; ++v) { cr[v] = br; ci[v] = bi; }
    for (int kc = 0; kc < NKC; ++kc) {
      const int d0 = 4 * kc + 2 * khalf;
      v2f wr, wi;
      wr[0] = w2r[d0 * BLK + 16 * j + row];
      wr[1] = w2r[(d0 + 1) * BLK + 16 * j + row];
      wi[0] = w2i[d0 * BLK + 16 * j + row];
      wi[1] = w2i[(d0 + 1) * BLK + 16 * j + row];
      cr = wmma4(a2r[kc], wr, cr);
      cr = wmma4(-a2i[kc], wi, cr);
      ci = wmma4(a2r[kc], wi, ci);
      ci = wmma4(a2i[kc], wr, ci);
    }
#pragma unroll
    for (int v = 0; v < 8; ++v) {
      const int M = khalf * 8 + v;
      sR[M * 97 + 16 * j + row] = sshrink(cr[v]);
      sI[M * 97 + 16 * j + row] = sshrink(ci[v]);
    }
  }

  for (int idx = lane; idx < BLK * 16; idx += 32) {
    const int col = idx >> 4;
    const int r   = idx & 15;
    const size_t g = xbase + (size_t)col * MPAD + m0 + r;
    Xr[g] = sR[r * 97 + col];
    Xi[g] = sI[r * 97 + col];
  }
}

// ---------------------------------------------------------------------------
// Kernel 4: inverse rfft (ortho) per (b,c); writes yT[b][c][0..8191].
// ---------------------------------------------------------------------------
__global__ __launch_bounds__(256) void k_fft_inv(
    const float* __restrict__ Yr, const float* __restrict__ Yi,
    float* __restrict__ yT) {
  __shared__ float2 bufA[N2];
  __shared__ float2 bufB[N2];
  const int bc  = blockIdx.x;
  const int tid = threadIdx.x;
  const size_t ib = (size_t)bc * MPAD;
  for (int k = tid; k < N2; k += 256) {
    const float xkr = Yr[ib + k],        xki = Yi[ib + k];
    const float xmr = Yr[ib + N2 - k],   xmi = Yi[ib + N2 - k];
    // Xe = (Xk + conj(Xm))/2 ; Xo = e^{+2pi i k/N} (Xk - conj(Xm))/2
    const float er = 0.5f * (xkr + xmr);
    const float ei = 0.5f * (xki - xmi);
    const float dr = 0.5f * (xkr - xmr);
    const float di = 0.5f * (xki + xmi);
    float th = 6.283185307179586f * (float)k / (float)SEQ;
    float sw, cw;
    __sincosf(th, &sw, &cw);
    const float xo_r = cw * dr - sw * di;
    const float xo_i = cw * di + sw * dr;
    float2 z;                 // Z' = Xe + i*Xo
    z.x = er - xo_i;
    z.y = ei + xo_r;
    bufA[k] = z;
  }

  fft4096(bufA, bufB, tid, +1.0f);

  const float sc = 0.022097086912079608f;  // sqrt(8192)/4096
  float2* op2 = (float2*)(yT + (size_t)bc * SEQ);
  for (int i = tid; i < N2; i += 256) {
    float2 z = bufA[i];
    float2 o;
    o.x = sc * z.x;  // y[2t]
    o.y = sc * z.y;  // y[2t+1]
    op2[i] = o;
  }
}

// ---------------------------------------------------------------------------
// Kernel 5: out[b,s,c] = yT[b,c,s] + x[b,s,c]
// ---------------------------------------------------------------------------
__global__ __launch_bounds__(256) void k_transpose_add(
    const float* __restrict__ yT, const float* __restrict__ x,
    float* __restrict__ out) {
  __shared__ float tile[32][33];
  const int b  = blockIdx.z;
  const int s0 = blockIdx.x * 32;
  const int c0 = blockIdx.y * 32;
  const int tx = threadIdx.x & 31;
  const int ty = threadIdx.x >> 5;
#pragma unroll
  for (int j = 0; j < 4; ++j) {
    const int c = c0 + ty + j * 8;
    tile[ty + j * 8][tx] = yT[((size_t)b * HIDDEN + c) * SEQ + (s0 + tx)];
  }
  __syncthreads();
#pragma unroll
  for (int j = 0; j < 4; ++j) {
    const size_t o = ((size_t)b * SEQ + (s0 + ty + j * 8)) * HIDDEN + (c0 + tx);
    out[o] = tile[tx][ty + j * 8] + x[o];
  }
}

// ---------------------------------------------------------------------------
extern "C" void kernel_launch(void* const* d_in, const int* in_sizes, int n_in,
                              void* d_out, int out_size, void* d_ws,
                              size_t ws_size, hipStream_t stream) {
  const float* x  = (const float*)d_in[0];
  const float* w1 = (const float*)d_in[1];
  const float* b1 = (const float*)d_in[2];
  const float* w2 = (const float*)d_in[3];
  const float* b2 = (const float*)d_in[4];
  float* out = (float*)d_out;

  float* ws = (float*)d_ws;
  const size_t XT_ELEMS = (size_t)BATCH * HIDDEN * SEQ;   // 25,165,824
  const size_t SP_ELEMS = (size_t)BATCH * HIDDEN * MPAD;  // 12,632,064
  float* xT = ws;                 // reused as yT after forward FFT consumed it
  float* Xr = ws + XT_ELEMS;
  float* Xi = Xr + SP_ELEMS;

  k_transpose_in<<<dim3(SEQ / 32, HIDDEN / 32, BATCH), 256, 0, stream>>>(x, xT);
  k_fft_fwd<<<dim3(BATCH * HIDDEN), 256, 0, stream>>>(xT, Xr, Xi);
  k_afno_mlp<<<dim3(65, NB, BATCH), 128, 0, stream>>>(Xr, Xi, w1, b1, w2, b2);
  k_fft_inv<<<dim3(BATCH * HIDDEN), 256, 0, stream>>>(Xr, Xi, xT);
  k_transpose_add<<<dim3(SEQ / 32, HIDDEN / 32, BATCH), 256, 0, stream>>>(
      xT, x, out);
}